// AutomatonPELayer_78512002171333
// MI455X (gfx1250) — compile-verified
//
#include <hip/hip_runtime.h>

// ---------------------------------------------------------------------------
// AutomatonPELayer on MI455X (gfx1250, wave32, WMMA).
//
// pe[t] = Q^t x0  (Q orthogonal 128x128),  out = pe @ W^T + b  (131072 x 512 f32)
//
// Log-depth power doubling kills the sequential scan; one HBM-write-bound
// f32-WMMA GEMM produces the 256 MB output:
//   t = a*4096 + s,  a in [0,32),  s in [0,4096)
//   PE0[:, s] = Q^s x0              (128 x 4096, 2 MB, L2 resident)
//   WP_a      = W @ Q^(4096*a)      (32 x 512 x 128, 8 MB, L2 resident)
//   out[t, e] = sum_k PE0[k,s] * WP_a[e,k] + b[e]
// All matmuls: V_WMMA_F32_16X16X4_F32, K=128 fully unrolled, leading dims are
// template constants so every load folds into an immediate global_load offset.
// ---------------------------------------------------------------------------

typedef __attribute__((ext_vector_type(2))) float v2f;
typedef __attribute__((ext_vector_type(8))) float v8f;

#define NSTATE 128
#define EMBED  512
#define SEQ    131072
#define CHUNK  4096
#define NBLK   (SEQ / CHUNK)   // 32

__device__ __forceinline__ v8f wmma_f32_16x16x4(v2f a, v2f b, v8f c) {
  // 8 args: (neg_a, A, neg_b, B, c_mod, C, reuse_a, reuse_b)
  return __builtin_amdgcn_wmma_f32_16x16x4_f32(false, a, false, b, (short)0, c,
                                               false, false);
}

// ---------------------------------------------------------------------------
// Seed: pe0 columns 0..15 via 15 sequential matvecs (single workgroup).
// PE0 layout: row-major [k][s], leading dim CHUNK.
// ---------------------------------------------------------------------------
__global__ void seed_kernel(const float* __restrict__ Q,
                            const float* __restrict__ x0,
                            float* __restrict__ PE0) {
  __shared__ float xs[NSTATE];
  const int i = threadIdx.x;            // 0..127
  float x = x0[i];
  xs[i] = x;
  PE0[i * CHUNK + 0] = x;
  __syncthreads();
  for (int s = 1; s < 16; ++s) {
    const float* row = Q + i * NSTATE;  // row i of Q (L0/L2 cached, 64 KB)
    float sum = 0.f;
#pragma unroll 8
    for (int j = 0; j < NSTATE; ++j) sum += row[j] * xs[j];
    __syncthreads();
    xs[i] = sum;
    PE0[i * CHUNK + s] = sum;
    __syncthreads();
  }
}

// ---------------------------------------------------------------------------
// Generic (optionally batched) f32 WMMA GEMM: C[z] = A[z] @ B, row-major,
// K = 128 hardcoded, leading dims as template constants so the fully unrolled
// K-loop uses only immediate load offsets. One wave per 16x16 output tile.
// grid = (N/16, M/16, batch).
//
// WMMA f32 16x16x4 lane layout (ISA 7.12.2):
//   A (16x4): lane l -> row = l&15, holds K = {k0, k0+1}, k0 = kk + 2*(l>>4)
//   B (4x16): lane l -> col = l&15, same K split
//   D: VGPR r -> row (tm + 8*(l>>4) + r), col = tn + (l&15)
// ---------------------------------------------------------------------------
template <int LDA, int LDB, int LDC>
__global__ void gemm_f32_wmma(const float* __restrict__ A,
                              const float* __restrict__ B,
                              float* __restrict__ C,
                              long long strideA, long long strideC) {
  const float* Ab = A + strideA * (long long)blockIdx.z;
  float* Cb = C + strideC * (long long)blockIdx.z;
  const int lane = threadIdx.x & 31;
  const int half = lane >> 4;
  const int l15  = lane & 15;
  const int koff = half * 2;
  const int tn = blockIdx.x * 16;
  const int tm = blockIdx.y * 16;

  // Per-lane base pointers; K-loop offsets are compile-time constants.
  const float* __restrict__ Abase = Ab + (tm + l15) * LDA + koff;  // A[row][koff]
  const float* __restrict__ Bbase = B + koff * LDB + tn + l15;     // B[koff][col]

  v8f acc = {0.f, 0.f, 0.f, 0.f, 0.f, 0.f, 0.f, 0.f};
#pragma unroll
  for (int kk = 0; kk < NSTATE; kk += 4) {
    v2f a = *(const v2f*)(Abase + kk);       // contiguous pair -> b64
    v2f b;
    b.x = Bbase[kk * LDB];                   // stride-LDB pair -> 2x b32
    b.y = Bbase[(kk + 1) * LDB];
    acc = wmma_f32_16x16x4(a, b, acc);
  }

  float* __restrict__ Crow = Cb + (tm + half * 8) * LDC + tn + l15;
#pragma unroll
  for (int r = 0; r < 8; ++r) Crow[r * LDC] = acc[r];
}

// ---------------------------------------------------------------------------
// WP_0 = W (plain copy into workspace).
// ---------------------------------------------------------------------------
__global__ void copy_kernel(const float* __restrict__ src,
                            float* __restrict__ dst, int n) {
  int i = blockIdx.x * blockDim.x + threadIdx.x;
  if (i < n) dst[i] = src[i];
}

// ---------------------------------------------------------------------------
// Final: Out[a*4096+s, e] = sum_k PE0[k,s] * WP_a[e,k] + b[e]
// A' = PE0^T (s x k) -> stride-CHUNK K pairs; B' = WP_a^T (k x e) -> contiguous
// K pairs (b64). 8 waves/block, all sharing the same WP_a 16x128 tile (L0).
// K=128 fully unrolled: 32 WMMAs, loads at immediate offsets only.
// grid = (EMBED/16, CHUNK/16/8, NBLK), block = 256 (8 wave32).
// ---------------------------------------------------------------------------
__global__ void out_kernel(const float* __restrict__ PE0,
                           const float* __restrict__ WP,
                           const float* __restrict__ bias,
                           float* __restrict__ Out) {
  const int lane = threadIdx.x & 31;
  const int wv   = threadIdx.x >> 5;
  const int half = lane >> 4;
  const int l15  = lane & 15;
  const int koff = half * 2;
  const int et = blockIdx.x * 16;                 // embed tile base
  const int st = (blockIdx.y * 8 + wv) * 16;      // time-in-chunk tile base
  const int a  = blockIdx.z;                      // chunk index

  // Per-lane base pointers; all K-loop offsets fold into immediates
  // (max A offset: 125*CHUNK*4B ~ 2 MB < 8 MB ioffset range).
  const float* __restrict__ Abase = PE0 + koff * CHUNK + st + l15;
  const float* __restrict__ Bbase =
      WP + (size_t)a * EMBED * NSTATE + (et + l15) * NSTATE + koff;

  v8f acc = {0.f, 0.f, 0.f, 0.f, 0.f, 0.f, 0.f, 0.f};
#pragma unroll
  for (int kk = 0; kk < NSTATE; kk += 4) {
    v2f av, bv;
    av.x = Abase[kk * CHUNK];                 // PE0^T pair: 2x b32
    av.y = Abase[(kk + 1) * CHUNK];
    bv = *(const v2f*)(Bbase + kk);           // WP_a^T pair: 1x b64
    acc = wmma_f32_16x16x4(av, bv, acc);
  }

  const float bb = bias[et + l15];
  float* __restrict__ Orow =
      Out + ((size_t)a * CHUNK + st + half * 8) * EMBED + et + l15;
#pragma unroll
  for (int r = 0; r < 8; ++r) Orow[(size_t)r * EMBED] = acc[r] + bb;
}

// ---------------------------------------------------------------------------
// Host orchestration (all on `stream`, graph-capture safe: no mallocs/syncs).
// ---------------------------------------------------------------------------
extern "C" void kernel_launch(void* const* d_in, const int* in_sizes, int n_in,
                              void* d_out, int out_size, void* d_ws,
                              size_t ws_size, hipStream_t stream) {
  (void)in_sizes; (void)n_in; (void)out_size; (void)ws_size;
  // dict order: sentence_len(i32), pos_initial(128), pos_transition(128x128),
  //             W(512x128), b(512)
  const float* x0 = (const float*)d_in[1];
  const float* Q  = (const float*)d_in[2];
  const float* W  = (const float*)d_in[3];
  const float* b  = (const float*)d_in[4];
  float* out = (float*)d_out;

  // Workspace layout (floats): qb0|qb1 ping-pong (2x16K), PE0 (512K), WP (2M)
  float* ws  = (float*)d_ws;
  float* qb0 = ws;
  float* qb1 = ws + 16384;
  float* PE0 = ws + 2 * 16384;
  float* WP  = PE0 + (size_t)NSTATE * CHUNK;    // 32 * 512 * 128 floats

  const dim3 wave(32);
  const dim3 g88(8, 8, 1);   // 128x128 output, 64 tiles

  // --- Phase 1: seed first 16 pe columns; build Q^16 by squaring ----------
  seed_kernel<<<dim3(1), dim3(128), 0, stream>>>(Q, x0, PE0);
  gemm_f32_wmma<128, 128, 128><<<g88, wave, 0, stream>>>(Q,   Q,   qb0, 0, 0); // Q^2
  gemm_f32_wmma<128, 128, 128><<<g88, wave, 0, stream>>>(qb0, qb0, qb1, 0, 0); // Q^4
  gemm_f32_wmma<128, 128, 128><<<g88, wave, 0, stream>>>(qb1, qb1, qb0, 0, 0); // Q^8
  gemm_f32_wmma<128, 128, 128><<<g88, wave, 0, stream>>>(qb0, qb0, qb1, 0, 0); // Q^16
  float* qcur = qb1;  // Q^16
  float* qalt = qb0;

  // --- Phase 2: PE doubling 16 -> 4096 columns (8 levels) ------------------
  int N = 16;
  for (int k = 0; k < 8; ++k) {
    // PE0[:, N:2N] = Qcur @ PE0[:, 0:N]
    gemm_f32_wmma<128, CHUNK, CHUNK>
        <<<dim3(N / 16, NSTATE / 16, 1), wave, 0, stream>>>(qcur, PE0, PE0 + N, 0, 0);
    // Qcur <- Qcur^2
    gemm_f32_wmma<128, 128, 128><<<g88, wave, 0, stream>>>(qcur, qcur, qalt, 0, 0);
    float* t = qcur; qcur = qalt; qalt = t;
    N *= 2;
  }
  // qcur == Q^4096

  // --- Phase 3: WP_a = W @ Q^(4096 a) by doubling on a (5 batched levels) --
  copy_kernel<<<dim3((EMBED * NSTATE + 255) / 256), dim3(256), 0, stream>>>(
      W, WP, EMBED * NSTATE);                        // WP_0 = W
  const long long wstride = (long long)EMBED * NSTATE;
  int m = 1;
  for (int j = 0; j < 5; ++j) {
    // WP_{m+z} = WP_z @ R, z in [0, m)
    gemm_f32_wmma<128, 128, 128>
        <<<dim3(NSTATE / 16, EMBED / 16, m), wave, 0, stream>>>(
            WP, qcur, WP + (size_t)m * wstride, wstride, wstride);
    if (j < 4) {  // R <- R^2
      gemm_f32_wmma<128, 128, 128><<<g88, wave, 0, stream>>>(qcur, qcur, qalt, 0, 0);
      float* t = qcur; qcur = qalt; qalt = t;
    }
    m *= 2;
  }

  // --- Phase 4: the fat GEMM (17.2 GFLOP, 256 MB write, HBM-bound) ---------
  out_kernel<<<dim3(EMBED / 16, CHUNK / 16 / 8, NBLK), dim3(256), 0, stream>>>(
      PE0, WP, b, out);
}